// GNNBlock_74285754352127
// MI455X (gfx1250) — compile-verified
//
#include <hip/hip_runtime.h>

typedef __attribute__((ext_vector_type(2))) float v2f;
typedef __attribute__((ext_vector_type(8))) float v8f;

#define FDIM 128
#define PAIRNORM_EPS 1e-5f

// ---------- init: out=0, deg=1 (self loop weight), stats=0 ----------
__global__ void k_init(float* __restrict__ out, float* __restrict__ deg,
                       float* __restrict__ colsum, float* __restrict__ sumsq,
                       int n_out, int n_nodes) {
    int i = blockIdx.x * blockDim.x + threadIdx.x;
    if (i < n_out)   out[i] = 0.0f;
    if (i < n_nodes) deg[i] = 1.0f;      // self-loop contributes weight 1
    if (i < FDIM)    colsum[i] = 0.0f;
    if (i == 0)      sumsq[0] = 0.0f;
}

// ---------- deg[dst] += w ----------
__global__ void k_deg(const int* __restrict__ dst, const float* __restrict__ w,
                      float* __restrict__ deg, int E) {
    int e = blockIdx.x * blockDim.x + threadIdx.x;
    if (e < E) atomicAdd(&deg[dst[e]], w[e]);
}

// ---------- deg -> deg^{-1/2} in place ----------
__global__ void k_dis(float* __restrict__ deg, int n) {
    int i = blockIdx.x * blockDim.x + threadIdx.x;
    if (i < n) {
        float d = deg[i];
        deg[i] = (d > 0.0f) ? (1.0f / sqrtf(d)) : 0.0f;
    }
}

// ---------- per-edge norm = dis[src]*w*dis[dst] ----------
__global__ void k_norm(const int* __restrict__ src, const int* __restrict__ dst,
                       const float* __restrict__ w, const float* __restrict__ dis,
                       float* __restrict__ norm, int E) {
    int e = blockIdx.x * blockDim.x + threadIdx.x;
    if (e < E) norm[e] = dis[src[e]] * w[e] * dis[dst[e]];
}

// ---------- h = x @ W via V_WMMA_F32_16X16X4_F32 ----------
// block = 256 (8 waves); block b covers rows [16b,16b+16), wave w covers cols [16w,16w+16)
__global__ void k_gemm(const float* __restrict__ x, const float* __restrict__ wmat,
                       float* __restrict__ h, int n_nodes) {
    const int lane = threadIdx.x & 31;
    const int wave = threadIdx.x >> 5;
    const int l16  = lane & 15;
    const int hi   = lane >> 4;            // 0: K pair {0,1}, 1: K pair {2,3}
    const int row0 = blockIdx.x * 16;
    const int col0 = wave * 16;
    if (row0 >= n_nodes) return;

    const float* arow = x + (size_t)(row0 + l16) * FDIM;   // A row for this lane
    v8f acc = {};
    for (int k = 0; k < FDIM; k += 4) {
        // A 16x4: lane(l16) = M, VGPRs hold K = 2*hi, 2*hi+1  (contiguous in row)
        v2f a = *(const v2f*)(arow + k + 2 * hi);
        // B 4x16: lane(l16) = N, VGPRs hold K = 2*hi, 2*hi+1  (strided columns of W)
        v2f b;
        b.x = wmat[(size_t)(k + 2 * hi    ) * FDIM + col0 + l16];
        b.y = wmat[(size_t)(k + 2 * hi + 1) * FDIM + col0 + l16];
        acc = __builtin_amdgcn_wmma_f32_16x16x4_f32(
                  false, a, false, b, (short)0, acc, false, false);
    }
    // D 16x16 f32: VGPR v, lane l -> M = v + 8*hi, N = l16
    float* hp = h + (size_t)row0 * FDIM + col0 + l16;
#pragma unroll
    for (int v = 0; v < 8; ++v)
        hp[(size_t)(v + 8 * hi) * FDIM] = acc[v];
}

// ---------- edge scatter: one wave per edge, float4 per lane ----------
__global__ void k_scatter(const float* __restrict__ h, const int* __restrict__ src,
                          const int* __restrict__ dst, const float* __restrict__ norm,
                          float* __restrict__ out, int E) {
    int e = blockIdx.x * 8 + (threadIdx.x >> 5);
    if (e >= E) return;
    int lane = threadIdx.x & 31;
    int s = src[e];
    int d = dst[e];
    float nm = norm[e];
    float4 v = ((const float4*)(h + (size_t)s * FDIM))[lane];
    float* op = out + (size_t)d * FDIM + lane * 4;
    atomicAdd(op + 0, v.x * nm);
    atomicAdd(op + 1, v.y * nm);
    atomicAdd(op + 2, v.z * nm);
    atomicAdd(op + 3, v.w * nm);
}

// ---------- add self-loop msg + bias, accumulate PairNorm stats ----------
#define ROWS_PER_BLOCK 64
__global__ void k_stats(float* __restrict__ out, const float* __restrict__ h,
                        const float* __restrict__ dis, const float* __restrict__ bias,
                        float* __restrict__ colsum, float* __restrict__ sumsq, int n) {
    const int f  = threadIdx.x;              // 0..127, one feature per thread
    const int n0 = blockIdx.x * ROWS_PER_BLOCK;
    const float b = bias[f];
    float cs = 0.0f, ss = 0.0f;
    for (int r = 0; r < ROWS_PER_BLOCK; ++r) {
        int node = n0 + r;
        if (node >= n) break;
        float dn = dis[node];
        float sn = dn * dn;                  // self-loop norm = dis^2 (w=1)
        size_t idx = (size_t)node * FDIM + f;
        float v = out[idx] + h[idx] * sn + b;
        out[idx] = v;
        cs += v;
        ss += v * v;
    }
    atomicAdd(&colsum[f], cs);               // thread owns feature f in this block
    // block-reduce ss (wave32 shuffle, then across 4 waves)
    for (int o = 16; o > 0; o >>= 1) ss += __shfl_xor(ss, o, 32);
    __shared__ float wss[4];
    if ((threadIdx.x & 31) == 0) wss[threadIdx.x >> 5] = ss;
    __syncthreads();
    if (threadIdx.x == 0)
        atomicAdd(sumsq, wss[0] + wss[1] + wss[2] + wss[3]);
}

// ---------- finalize stats: mean[f], inv_denom ----------
__global__ void k_mean(const float* __restrict__ colsum, const float* __restrict__ sumsq,
                       float* __restrict__ mean, float* __restrict__ invd, int n) {
    const int f = threadIdx.x;               // 128 threads
    float m = colsum[f] / (float)n;
    mean[f] = m;
    float s = m * m;
    for (int o = 16; o > 0; o >>= 1) s += __shfl_xor(s, o, 32);
    __shared__ float wss[4];
    if ((f & 31) == 0) wss[f >> 5] = s;
    __syncthreads();
    if (f == 0) {
        float meansq = wss[0] + wss[1] + wss[2] + wss[3];       // sum_f mean_f^2
        // sum||c||^2 / N = sumsq/N - sum_f mean_f^2
        float var = sumsq[0] / (float)n - meansq;
        invd[0] = 1.0f / sqrtf(PAIRNORM_EPS + var);
    }
}

// ---------- center, scale, ReLU ----------
__global__ void k_final(float* __restrict__ out, const float* __restrict__ mean,
                        const float* __restrict__ invd, int total) {
    int i = blockIdx.x * blockDim.x + threadIdx.x;
    if (i < total) {
        float v = (out[i] - mean[i & (FDIM - 1)]) * invd[0];   // PAIRNORM_SCALE == 1
        out[i] = fmaxf(v, 0.0f);
    }
}

extern "C" void kernel_launch(void* const* d_in, const int* in_sizes, int n_in,
                              void* d_out, int out_size, void* d_ws, size_t ws_size,
                              hipStream_t stream) {
    const float* x    = (const float*)d_in[0];   // [N,128]
    const int*   ei   = (const int*)d_in[1];     // [2,E] flat: src then dst
    const float* ew   = (const float*)d_in[2];   // [E]
    const float* wmat = (const float*)d_in[3];   // [128,128] (in,out)
    const float* bias = (const float*)d_in[4];   // [128]
    float* out = (float*)d_out;

    const int N = in_sizes[0] / FDIM;
    const int E = in_sizes[2];
    const int* src = ei;
    const int* dst = ei + E;

    // workspace carve-out (256B aligned): h[N*128], dis[N], norm[E], stats
    char* ws = (char*)d_ws;
    size_t off = 0;
    auto alignup = [](size_t v) { return (v + 255) & ~(size_t)255; };
    float* h      = (float*)(ws + off); off = alignup(off + (size_t)N * FDIM * sizeof(float));
    float* dis    = (float*)(ws + off); off = alignup(off + (size_t)N * sizeof(float));   // deg then dis
    float* norm   = (float*)(ws + off); off = alignup(off + (size_t)E * sizeof(float));
    float* colsum = (float*)(ws + off); off = alignup(off + FDIM * sizeof(float));
    float* sumsq  = (float*)(ws + off); off = alignup(off + sizeof(float));
    float* mean   = (float*)(ws + off); off = alignup(off + FDIM * sizeof(float));
    float* invd   = (float*)(ws + off); off = alignup(off + sizeof(float));
    (void)off; (void)ws_size; (void)n_in; (void)out_size;

    const int total = N * FDIM;

    k_init   <<<(total + 255) / 256, 256, 0, stream>>>(out, dis, colsum, sumsq, total, N);
    k_deg    <<<(E + 255) / 256,     256, 0, stream>>>(dst, ew, dis, E);
    k_dis    <<<(N + 255) / 256,     256, 0, stream>>>(dis, N);
    k_gemm   <<<(N + 15) / 16,       256, 0, stream>>>(x, wmat, h, N);
    k_norm   <<<(E + 255) / 256,     256, 0, stream>>>(src, dst, ew, dis, norm, E);
    k_scatter<<<(E + 7) / 8,         256, 0, stream>>>(h, src, dst, norm, out, E);
    k_stats  <<<(N + ROWS_PER_BLOCK - 1) / ROWS_PER_BLOCK, FDIM, 0, stream>>>(out, h, dis, bias, colsum, sumsq, N);
    k_mean   <<<1, FDIM, 0, stream>>>(colsum, sumsq, mean, invd, N);
    k_final  <<<(total + 255) / 256, 256, 0, stream>>>(out, mean, invd, total);
}